// SpModel_326417515069
// MI455X (gfx1250) — compile-verified
//
#include <hip/hip_runtime.h>
#include <cstdint>

#define G_ 128
#define N_ 32
#define H_ 128
#define L_ 6

typedef __attribute__((ext_vector_type(2))) float v2f;
typedef __attribute__((ext_vector_type(8))) float v8f;

// ---------------------------------------------------------------------------
// CDNA5 async global<->LDS helpers (ASYNCcnt-tracked, CDNA5 ISA)
// VDST/VSRC VGPR holds the per-lane LDS byte offset -> free scatter/transpose.
// ---------------------------------------------------------------------------
__device__ __forceinline__ void async_ld_b128(uint32_t lds_off, const void* g) {
    asm volatile("global_load_async_to_lds_b128 %0, %1, off"
                 :: "v"(lds_off), "v"(g) : "memory");
}
__device__ __forceinline__ void async_ld_b32(uint32_t lds_off, const void* g) {
    asm volatile("global_load_async_to_lds_b32 %0, %1, off"
                 :: "v"(lds_off), "v"(g) : "memory");
}
__device__ __forceinline__ void async_st_b32(void* g, uint32_t lds_off) {
    asm volatile("global_store_async_from_lds_b32 %0, %1, off"
                 :: "v"(g), "v"(lds_off) : "memory");
}
__device__ __forceinline__ void wait_async0() {
    asm volatile("s_wait_asynccnt 0" ::: "memory");
}
// low 32 bits of a flat pointer to __shared__ = LDS byte offset (aperture form)
__device__ __forceinline__ uint32_t lds_off(const void* p) {
    return (uint32_t)(uintptr_t)p;
}

// ---------------------------------------------------------------------------
// K1: x = emb_x[x_idx]; x0 = x@W0+b0 ; x1 = x@W1+b1   (one block per row)
// ---------------------------------------------------------------------------
__global__ __launch_bounds__(128) void k_encode(
    const int* __restrict__ x_idx, const float* __restrict__ emb_x,
    const float* __restrict__ W0, const float* __restrict__ b0,
    const float* __restrict__ W1, const float* __restrict__ b1,
    float* __restrict__ x0, float* __restrict__ x1)
{
    __shared__ float xr[H_];
    const int r = blockIdx.x, h = threadIdx.x;
    xr[h] = emb_x[x_idx[r] * H_ + h];
    __syncthreads();
    float a0 = b0[h], a1 = b1[h];
    for (int k = 0; k < H_; ++k) {
        const float xv = xr[k];
        a0 = fmaf(xv, W0[k * H_ + h], a0);
        a1 = fmaf(xv, W1[k * H_ + h], a1);
    }
    x0[r * H_ + h] = a0;
    x1[r * H_ + h] = a1;
}

// ---------------------------------------------------------------------------
// K2: X[g,i,j,h] = x0[g,i,h]*x1[g,j,h]*emb_tf[tf_idx];  Aval = emb_ea[ea]*adj
// ---------------------------------------------------------------------------
__global__ __launch_bounds__(128) void k_build(
    const float* __restrict__ x0, const float* __restrict__ x1,
    const int* __restrict__ tf_idx, const float* __restrict__ emb_tf,
    const int* __restrict__ ea_idx, const float* __restrict__ emb_ea,
    const int* __restrict__ adj,
    float* __restrict__ X, float* __restrict__ Aval)
{
    const int p = blockIdx.x;            // (g*N + i)*N + j
    const int h = threadIdx.x;
    const int j  = p & (N_ - 1);
    const int gi = p >> 5;               // g*N + i
    const int g  = gi >> 5;
    const float v = x0[gi * H_ + h] * x1[(g * N_ + j) * H_ + h] *
                    emb_tf[tf_idx[p] * H_ + h];
    X[p * H_ + h]    = v;
    Aval[p * H_ + h] = emb_ea[ea_idx[p] * H_ + h] * (float)adj[p];
}

// ---------------------------------------------------------------------------
// K3: tX = relu(LN(X @ Wc[l] + bc[l]))   — WMMA f32 GEMM, fused LN epilogue.
// Block = 256 thr (8 waves). Tile: 64 rows x 128 cols. Grid = 2048.
// Async-LDS staging: X tile row-major (b128), W transposed Wt[n][k] (b32,
// per-lane scatter). Wt stride 129 -> contiguous b64 B-fragments, bank-free.
// ---------------------------------------------------------------------------
__global__ __launch_bounds__(256) void k_conv_gemm_ln(
    const float* __restrict__ X, const float* __restrict__ Wc_l,
    const float* __restrict__ bc_l, const float* __restrict__ gc_l,
    const float* __restrict__ cc_l, float* __restrict__ tX)
{
    __shared__ float lds[64 * 132 + 128 * 129];
    float* Xs = lds;                 // 64 x 132
    float* Wt = lds + 64 * 132;      // 128(n) x 129 (k contiguous)
    float* Os = lds;                 // aliases Xs after the K-loop barrier

    const int tid  = threadIdx.x;
    const int lane = tid & 31;
    const int wave = tid >> 5;
    const int row0 = blockIdx.x * 64;

    // async load 64x128 X tile (b128, coalesced, direct-to-LDS)
    const uint32_t xs_b = lds_off(Xs);
    for (int q = 0; q < 8; ++q) {
        const int vi = q * 256 + tid;
        const int r = vi >> 5, c = (vi & 31) * 4;
        async_ld_b128(xs_b + (uint32_t)(r * 132 + c) * 4u,
                      X + (row0 + r) * H_ + c);
    }
    // async transpose-load W: global row-major, LDS Wt[n*129 + k]
    const uint32_t wt_b = lds_off(Wt);
    for (int q = 0; q < 64; ++q) {
        const int t = wave * 64 + q;          // uniform per wave
        const int k = t & 127, nb = t >> 7;
        const int n = nb * 32 + lane;         // 32 consecutive n -> 128B line
        async_ld_b32(wt_b + (uint32_t)(n * 129 + k) * 4u, Wc_l + k * H_ + n);
    }
    wait_async0();
    __syncthreads();

    const int rt  = wave >> 1;        // row tile 0..3 (16 rows each)
    const int ct0 = (wave & 1) * 4;   // 4 col tiles per wave
    const int m   = lane & 15;
    const int ko  = (lane >> 4) * 2;  // lanes 16-31 hold K+2,K+3
    const int ro  = (lane >> 4) * 8;  // C/D row offset for high lanes

    v8f acc[4];
    for (int t = 0; t < 4; ++t) {     // bias folded into accumulator init
        const float cv = bc_l[(ct0 + t) * 16 + m];
        for (int v = 0; v < 8; ++v) acc[t][v] = cv;
    }
    for (int s = 0; s < 32; ++s) {    // K = 128 in steps of 4
        v2f a;
        const float* ap = &Xs[(rt * 16 + m) * 132 + s * 4 + ko];
        a.x = ap[0]; a.y = ap[1];
        for (int t = 0; t < 4; ++t) {
            const float* wp = &Wt[((ct0 + t) * 16 + m) * 129 + s * 4 + ko];
            v2f b;
            b.x = wp[0]; b.y = wp[1];           // contiguous -> ds_load_b64
            acc[t] = __builtin_amdgcn_wmma_f32_16x16x4_f32(
                false, a, false, b, (short)0, acc[t], false, false);
        }
    }
    __syncthreads();                  // all Xs reads done before Os aliasing
    for (int t = 0; t < 4; ++t)
        for (int v = 0; v < 8; ++v)
            Os[(rt * 16 + ro + v) * 132 + (ct0 + t) * 16 + m] = acc[t][v];
    __syncthreads();

    // LayerNorm + ReLU: each wave handles 8 rows, shuffle-reduce over H=128
    for (int rr = 0; rr < 8; ++rr) {
        const int r = wave * 8 + rr;
        float s = 0.f, ss = 0.f, vals[4];
        for (int q = 0; q < 4; ++q) {
            const float v = Os[r * 132 + lane + q * 32];
            vals[q] = v; s += v; ss += v * v;
        }
        for (int o = 16; o > 0; o >>= 1) {
            s  += __shfl_xor(s,  o, 32);
            ss += __shfl_xor(ss, o, 32);
        }
        const float mean = s * (1.f / 128.f);
        const float var  = ss * (1.f / 128.f) - mean * mean;
        const float rs   = rsqrtf(var + 1e-5f);
        float* orow = tX + (row0 + r) * H_;
        for (int q = 0; q < 4; ++q) {
            const int c = lane + q * 32;
            const float y = (vals[q] - mean) * rs * gc_l[c] + cc_l[c];
            orow[c] = fmaxf(y, 0.f);
        }
    }
}

// ---------------------------------------------------------------------------
// K4: X[g,:,:,h] += tX[g,:,:,h] @ Aval[g,:,:,h]  (per-h 32x32x32, WMMA f32)
// Block = (graph g, 8-channel slice hb). 8 waves, one h channel each.
// Residual free: C fragment initialized from the X tile.
// Async b32 loads do the h-transpose for free (per-lane LDS scatter);
// Aval stored K-transposed (AsT[j][k]) so B fragments are b64.
// Slice stride 1060 + row stride 33 -> 32-distinct-bank transpose stores.
// ---------------------------------------------------------------------------
__global__ __launch_bounds__(256) void k_msg_wmma(
    const float* __restrict__ tX, const float* __restrict__ Aval,
    float* __restrict__ X)
{
    __shared__ float tXs[8 * 1060];   // [h'][i*33 + k]
    __shared__ float AsT[8 * 1060];   // [h'][j*33 + k]  (K-transposed)
    __shared__ float Xs2[8 * 1060];   // [h'][i*33 + j]

    const int tid  = threadIdx.x;
    const int lane = tid & 31;
    const int wave = tid >> 5;
    const int g  = blockIdx.x >> 4;
    const int hb = (blockIdx.x & 15) * 8;
    const int gbase = g * N_ * N_ * H_;

    const uint32_t tXs_b = lds_off(tXs);
    const uint32_t AsT_b = lds_off(AsT);
    const uint32_t Xs2_b = lds_off(Xs2);

    for (int q = 0; q < 32; ++q) {
        const int f  = q * 256 + tid;
        const int hp = f & 7, aa = (f >> 3) & 31, bb = f >> 8;
        const int ga = gbase + (bb * N_ + aa) * H_ + hb + hp;
        async_ld_b32(tXs_b + (uint32_t)(hp * 1060 + bb * 33 + aa) * 4u, tX + ga);   // [i=bb][k=aa]
        async_ld_b32(AsT_b + (uint32_t)(hp * 1060 + aa * 33 + bb) * 4u, Aval + ga); // [j=aa][k=bb]
        async_ld_b32(Xs2_b + (uint32_t)(hp * 1060 + bb * 33 + aa) * 4u, X + ga);    // [i=bb][j=aa]
    }
    wait_async0();
    __syncthreads();

    const float* tA = &tXs[wave * 1060];
    const float* tB = &AsT[wave * 1060];
    float*       tC = &Xs2[wave * 1060];
    const int m  = lane & 15;
    const int ko = (lane >> 4) * 2;
    const int ro = (lane >> 4) * 8;

    for (int it = 0; it < 32; it += 16)
    for (int jt = 0; jt < 32; jt += 16) {
        v8f acc;
        for (int v = 0; v < 8; ++v)               // C := X tile (residual)
            acc[v] = tC[(it + ro + v) * 33 + jt + m];
        for (int s = 0; s < 8; ++s) {             // K = 32 in steps of 4
            v2f a, b;
            const float* ap = &tA[(it + m) * 33 + s * 4 + ko];
            a.x = ap[0]; a.y = ap[1];             // b64
            const float* bp = &tB[(jt + m) * 33 + s * 4 + ko];
            b.x = bp[0]; b.y = bp[1];             // b64 (K-transposed layout)
            acc = __builtin_amdgcn_wmma_f32_16x16x4_f32(
                false, a, false, b, (short)0, acc, false, false);
        }
        for (int v = 0; v < 8; ++v)
            tC[(it + ro + v) * 33 + jt + m] = acc[v];
    }
    __syncthreads();

    // async write-back from LDS (de-transpose via per-lane LDS source offsets)
    for (int q = 0; q < 32; ++q) {
        const int f  = q * 256 + tid;
        const int hp = f & 7, aa = (f >> 3) & 31, bb = f >> 8;
        async_st_b32(X + gbase + (bb * N_ + aa) * H_ + hb + hp,
                     Xs2_b + (uint32_t)(hp * 1060 + bb * 33 + aa) * 4u);
    }
    wait_async0();
}

// ---------------------------------------------------------------------------
// K5a: xs[g,i,h] = mean_j X[g,i,j,h]
// ---------------------------------------------------------------------------
__global__ __launch_bounds__(128) void k_poolmean(
    const float* __restrict__ X, float* __restrict__ xs)
{
    const int r = blockIdx.x;      // g*N + i
    const int h = threadIdx.x;
    const float* base = X + r * N_ * H_ + h;
    float s = 0.f;
    for (int j = 0; j < N_; ++j) s += base[j * H_];
    xs[r * H_ + h] = s * (1.f / (float)N_);
}

// ---------------------------------------------------------------------------
// K5b: out = relu(LN(in @ W + b, gamma, beta))   (one block per row)
// ---------------------------------------------------------------------------
__global__ __launch_bounds__(128) void k_rowmlp_ln_relu(
    const float* __restrict__ in, const float* __restrict__ W,
    const float* __restrict__ bias, const float* __restrict__ gamma,
    const float* __restrict__ beta, float* __restrict__ out)
{
    __shared__ float row[H_];
    __shared__ float buf[H_];
    const int r = blockIdx.x, h = threadIdx.x;
    row[h] = in[r * H_ + h];
    __syncthreads();
    float a = bias[h];
    for (int k = 0; k < H_; ++k) a = fmaf(row[k], W[k * H_ + h], a);
    buf[h] = a; __syncthreads();
    for (int o = 64; o > 0; o >>= 1) { if (h < o) buf[h] += buf[h + o]; __syncthreads(); }
    const float mean = buf[0] * (1.f / 128.f);
    __syncthreads();
    const float d = a - mean;
    buf[h] = d * d; __syncthreads();
    for (int o = 64; o > 0; o >>= 1) { if (h < o) buf[h] += buf[h + o]; __syncthreads(); }
    const float var = buf[0] * (1.f / 128.f);
    const float y = d * rsqrtf(var + 1e-5f) * gamma[h] + beta[h];
    out[r * H_ + h] = fmaxf(y, 0.f);
}

// ---------------------------------------------------------------------------
// K5c: hg[g,h] = sum_i ys[g,i,h]
// ---------------------------------------------------------------------------
__global__ __launch_bounds__(128) void k_graphsum(
    const float* __restrict__ ys, float* __restrict__ hg)
{
    const int g = blockIdx.x, h = threadIdx.x;
    float s = 0.f;
    for (int i = 0; i < N_; ++i) s += ys[(g * N_ + i) * H_ + h];
    hg[g * H_ + h] = s;
}

// ---------------------------------------------------------------------------
// K5d: out[g] = h2[g,:] . Wq2 + bq2
// ---------------------------------------------------------------------------
__global__ __launch_bounds__(128) void k_final(
    const float* __restrict__ h2, const float* __restrict__ Wq2,
    const float* __restrict__ bq2, float* __restrict__ out)
{
    __shared__ float buf[H_];
    const int g = blockIdx.x, h = threadIdx.x;
    buf[h] = h2[g * H_ + h] * Wq2[h];
    __syncthreads();
    for (int o = 64; o > 0; o >>= 1) { if (h < o) buf[h] += buf[h + o]; __syncthreads(); }
    if (h == 0) out[g] = buf[0] + bq2[0];
}

// ---------------------------------------------------------------------------
extern "C" void kernel_launch(void* const* d_in, const int* in_sizes, int n_in,
                              void* d_out, int out_size, void* d_ws, size_t ws_size,
                              hipStream_t stream)
{
    (void)in_sizes; (void)n_in; (void)out_size; (void)ws_size;

    const int*   x_idx  = (const int*)d_in[0];
    const int*   ea_idx = (const int*)d_in[1];
    const int*   tf_idx = (const int*)d_in[2];
    const int*   adj    = (const int*)d_in[3];
    const float* emb_x  = (const float*)d_in[4];
    const float* emb_ea = (const float*)d_in[5];
    const float* emb_tf = (const float*)d_in[6];
    const float* W0  = (const float*)d_in[7];
    const float* b0  = (const float*)d_in[8];
    const float* W1  = (const float*)d_in[9];
    const float* b1  = (const float*)d_in[10];
    const float* Wc  = (const float*)d_in[11];
    const float* bc  = (const float*)d_in[12];
    const float* gc  = (const float*)d_in[13];
    const float* cc  = (const float*)d_in[14];
    const float* Wp  = (const float*)d_in[15];
    const float* bp  = (const float*)d_in[16];
    const float* gp  = (const float*)d_in[17];
    const float* cp  = (const float*)d_in[18];
    const float* Wq1 = (const float*)d_in[19];
    const float* bq1 = (const float*)d_in[20];
    const float* gq1 = (const float*)d_in[21];
    const float* cq1 = (const float*)d_in[22];
    const float* Wq2 = (const float*)d_in[23];
    const float* bq2 = (const float*)d_in[24];
    float* out = (float*)d_out;

    // workspace layout (floats)
    const size_t R  = (size_t)G_ * N_;            // 4096 rows
    const size_t T  = (size_t)G_ * N_ * N_ * H_;  // 16,777,216 (X/Aval/tX)
    float* ws   = (float*)d_ws;
    float* x0   = ws;                 // R*H
    float* x1   = x0 + R * H_;
    float* X    = x1 + R * H_;        // T
    float* Aval = X + T;              // T
    float* tXb  = Aval + T;           // T
    float* xs   = tXb + T;            // R*H
    float* ys   = xs + R * H_;        // R*H
    float* hg   = ys + R * H_;        // G*H
    float* h2   = hg + (size_t)G_ * H_;

    k_encode<<<dim3((unsigned)R), dim3(H_), 0, stream>>>(
        x_idx, emb_x, W0, b0, W1, b1, x0, x1);

    k_build<<<dim3((unsigned)(G_ * N_ * N_)), dim3(H_), 0, stream>>>(
        x0, x1, tf_idx, emb_tf, ea_idx, emb_ea, adj, X, Aval);

    const unsigned gemm_blocks = (G_ * N_ * N_) / 64;   // 2048
    const unsigned msg_blocks  = G_ * (H_ / 8);         // 2048
    for (int l = 0; l < L_; ++l) {
        k_conv_gemm_ln<<<dim3(gemm_blocks), dim3(256), 0, stream>>>(
            X, Wc + (size_t)l * H_ * H_, bc + l * H_, gc + l * H_, cc + l * H_, tXb);
        k_msg_wmma<<<dim3(msg_blocks), dim3(256), 0, stream>>>(tXb, Aval, X);
    }

    k_poolmean<<<dim3((unsigned)R), dim3(H_), 0, stream>>>(X, xs);
    k_rowmlp_ln_relu<<<dim3((unsigned)R), dim3(H_), 0, stream>>>(xs, Wp, bp, gp, cp, ys);
    k_graphsum<<<dim3(G_), dim3(H_), 0, stream>>>(ys, hg);
    k_rowmlp_ln_relu<<<dim3(G_), dim3(H_), 0, stream>>>(hg, Wq1, bq1, gq1, cq1, h2);
    k_final<<<dim3(G_), dim3(H_), 0, stream>>>(h2, Wq2, bq2, out);
}